// Top2Router_66116726554789
// MI455X (gfx1250) — compile-verified
//
#include <hip/hip_runtime.h>
#include <math.h>

// Top-2 MoE router for MI455X (gfx1250).
// logits = x @ W^T + b ; softmax over E=8 ; top-2.
// Memory-bound: 134 MB of x @ 23.3 TB/s => ~5.8us floor. Compute via
// V_WMMA_F32_16X16X4_F32 (A = 16 tokens x 4K of x, B = 4K x 16 experts of W^T,
// experts 8..15 duplicated -> outputs ignored).

typedef __attribute__((ext_vector_type(2))) float v2f;
typedef __attribute__((ext_vector_type(8))) float v8f;

#define B_  4
#define S_  4096
#define D_  2048
#define E_  8
#define T_  (B_ * S_)              // 16384 tokens
#define WAVES_PER_WG  4
#define TOK_PER_WAVE  16
#define TOK_PER_WG    (WAVES_PER_WG * TOK_PER_WAVE)   // 64
#define LDS_STRIDE    9            // 9*t mod 64 distinct for t in [0,32) -> conflict-free

__global__ __launch_bounds__(128) void top2router_kernel(
    const float* __restrict__ x,     // [T_, D_]
    const float* __restrict__ Wm,    // [E_, D_]
    const float* __restrict__ bias,  // [E_]
    float* __restrict__ out)         // [2T | 2T | 8T] = top2_val, top2_idx, gate
{
    __shared__ float lds_logits[TOK_PER_WG * LDS_STRIDE];

    const int tid  = threadIdx.x;
    const int wave = tid >> 5;
    const int lane = tid & 31;
    const int half = lane >> 4;      // 0: K+{0,1} / M rows 0..7 ; 1: K+{2,3} / M rows 8..15
    const int m    = lane & 15;

    const int tok_base = blockIdx.x * TOK_PER_WG + wave * TOK_PER_WAVE;

    // A fragment source: x row (token tok_base+m), contiguous float2 at k0 + 2*half.
    const float* xrow = x  + (size_t)(tok_base + m) * D_ + 2 * half;
    // B fragment source: W^T column n = expert (lane&15); clamp to &7 (duplicate
    // experts in padding columns 8..15 -> those output columns are never read).
    const float* wrow = Wm + (size_t)(lane & 7)    * D_ + 2 * half;

    v8f acc = {};
    for (int kb = 0; kb < D_; kb += 128) {
        __builtin_prefetch(xrow + kb + 1024, 0, 3);   // global_prefetch_b8, 4KB ahead
#pragma unroll
        for (int kk = 0; kk < 128; kk += 4) {
            v2f a = *(const v2f*)(xrow + kb + kk);
            v2f w = *(const v2f*)(wrow + kb + kk);
            // 8 args: (neg_a, A, neg_b, B, c_mod, C, reuse_a, reuse_b)
            acc = __builtin_amdgcn_wmma_f32_16x16x4_f32(
                false, a, false, w, (short)0, acc, false, false);
        }
    }

    // D layout: lane -> column N = expert = lane&15 ; VGPR v -> row M = v + 8*half.
    if (m < E_) {
#pragma unroll
        for (int v = 0; v < 8; ++v) {
            int tloc = wave * TOK_PER_WAVE + v + 8 * half;
            lds_logits[tloc * LDS_STRIDE + m] = acc[v];
        }
    }
    __syncthreads();

    // Phase 2: one thread per token -> bias, softmax, top-2, outputs.
    if (tid < TOK_PER_WG) {
        const int t = blockIdx.x * TOK_PER_WG + tid;

        float lg[E_];
        float mx = -INFINITY;
#pragma unroll
        for (int e = 0; e < E_; ++e) {
            lg[e] = lds_logits[tid * LDS_STRIDE + e] + bias[e];
            mx = fmaxf(mx, lg[e]);
        }
        float sum = 0.f;
#pragma unroll
        for (int e = 0; e < E_; ++e) { lg[e] = __expf(lg[e] - mx); sum += lg[e]; }
        const float inv = 1.f / sum;

        float* out_val  = out;
        float* out_idx  = out + 2 * (size_t)T_;
        float* out_gate = out + 4 * (size_t)T_;

        float g[E_];
#pragma unroll
        for (int e = 0; e < E_; ++e) {
            g[e] = lg[e] * inv;
            out_gate[(size_t)t * E_ + e] = g[e];
        }

        int i1 = 0; float v1 = g[0];
#pragma unroll
        for (int e = 1; e < E_; ++e) if (g[e] > v1) { v1 = g[e]; i1 = e; }
        int i2 = -1; float v2 = -INFINITY;
#pragma unroll
        for (int e = 0; e < E_; ++e) if (e != i1 && g[e] > v2) { v2 = g[e]; i2 = e; }

        out_val[2 * (size_t)t + 0] = v1;
        out_val[2 * (size_t)t + 1] = v2;
        out_idx[2 * (size_t)t + 0] = (float)i1;
        out_idx[2 * (size_t)t + 1] = (float)i2;
    }
}

extern "C" void kernel_launch(void* const* d_in, const int* in_sizes, int n_in,
                              void* d_out, int out_size, void* d_ws, size_t ws_size,
                              hipStream_t stream) {
    (void)in_sizes; (void)n_in; (void)d_ws; (void)ws_size; (void)out_size;
    const float* x    = (const float*)d_in[0];
    const float* Wm   = (const float*)d_in[1];
    const float* bias = (const float*)d_in[2];
    float*       out  = (float*)d_out;

    dim3 grid(T_ / TOK_PER_WG);   // 256 workgroups
    dim3 block(128);              // 4 waves (wave32)
    hipLaunchKernelGGL(top2router_kernel, grid, block, 0, stream, x, Wm, bias, out);
}